// Repro_78726750535795
// MI455X (gfx1250) — compile-verified
//
#include <hip/hip_runtime.h>
#include <stdint.h>

// -------------------------------------------------------------------------
// Bilinear 2x downsample (exact 2:1 ratio => uniform 0.25-weight 2x2 box
// average) for (8,3,2048,2048) fp32 -> (8,3,1024,1024) round-to-u8 values.
//
// Memory-bound (~504 MB total @ 23.3 TB/s => ~22 us floor). Data movement is
// done with the CDNA5 Tensor Data Mover: one 2048x2-fp32 tile (16 KB) DMA'd
// into LDS per workgroup, synchronized with TENSORcnt, consumed by 8 wave32s
// via 128-bit LDS loads and written back with 128-bit global stores.
// -------------------------------------------------------------------------

typedef __attribute__((ext_vector_type(4))) unsigned int v4u;
typedef __attribute__((ext_vector_type(8))) int          v8i;
typedef __attribute__((ext_vector_type(4))) int          v4i;
typedef __attribute__((ext_vector_type(4))) float        f4;

#define IN_H  2048
#define IN_W  2048
#define OUT_H 1024
#define OUT_W 1024

__global__ __launch_bounds__(256) void downsample2x_tdm_kernel(
    const float* __restrict__ src,   // (NC, IN_H, IN_W)
    float* __restrict__ dst)         // (NC, OUT_H, OUT_W), rounded u8 values
{
    __shared__ float lds[2 * IN_W];  // two source rows, 16 KB

    const uint32_t oy = blockIdx.x;   // output row
    const uint32_t nc = blockIdx.y;   // fused N*C index
    const uint32_t t  = threadIdx.x;  // 0..255

    // ---- TDM: DMA source rows {2*oy, 2*oy+1} into LDS -------------------
    // Tensor instructions ignore EXEC, so gate by a wave-uniform branch:
    // only wave 0 (t<32) reaches the issue point.
    if (t < 32) {
        const uint64_t gaddr =
            (uint64_t)(uintptr_t)(src + ((size_t)nc * IN_H + (size_t)2 * oy) * IN_W);
        const uint32_t lds_addr = (uint32_t)(uintptr_t)(&lds[0]);

        // D# group 0 (128b): count=1 | lds_addr | global_addr[56:0] | type=2
        v4u g0;
        g0.x = 1u;                                   // count=1, user mode
        g0.y = lds_addr;                             // LDS byte address
        g0.z = (uint32_t)gaddr;                      // global_addr[31:0]
        g0.w = ((uint32_t)(gaddr >> 32) & 0x01FFFFFFu) | (2u << 30); // [56:32] | type=2

        // D# group 1 (256b): 2-D tile, fp32 elements
        //  data_size=2 (4B)   tensor_dim0=IN_W  tensor_dim1=2
        //  tile_dim0=IN_W     tile_dim1=2       tile_dim2=0 (2-D)
        //  tensor_dim0_stride=IN_W  tensor_dim1_stride=IN_W
        v8i g1;
        g1[0] = (int)(2u << 16);                                 // data_size=4B, wg_mask=0
        g1[1] = (int)(((uint32_t)IN_W & 0xFFFFu) << 16);         // tensor_dim0[15:0]
        g1[2] = (int)(((uint32_t)IN_W >> 16) | (2u << 16));      // tensor_dim0[31:16], tensor_dim1=2
        g1[3] = (int)(((uint32_t)IN_W & 0xFFFFu) << 16);         // tile_dim0=IN_W
        g1[4] = (int)2;                                          // tile_dim1=2, tile_dim2=0
        g1[5] = (int)IN_W;                                       // tensor_dim0_stride[31:0]
        g1[6] = (int)(((uint32_t)IN_W & 0xFFFFu) << 16);         // stride0[47:32]=0, stride1[15:0]=IN_W
        g1[7] = 0;                                               // stride1[47:16]=0

        // Groups 2/3: benign for a 2-D tile (tile_dim3/4 = 0)
        v4i g2; g2[0] = 1; g2[1] = 0; g2[2] = 0; g2[3] = 0;      // tensor_dim2=1
        v4i g3; g3[0] = 0; g3[1] = 0; g3[2] = 0; g3[3] = 0;

        // Extra descriptor group required by the 6-arg builtin on this
        // toolchain (clang-23 / therock-10.0); zero-filled — unused for a
        // 2-D tile.
        v8i g4;
        g4[0] = 0; g4[1] = 0; g4[2] = 0; g4[3] = 0;
        g4[4] = 0; g4[5] = 0; g4[6] = 0; g4[7] = 0;

        __builtin_amdgcn_tensor_load_to_lds(g0, g1, g2, g3, g4, /*cpol=*/0);
        __builtin_amdgcn_s_wait_tensorcnt(0);
    }
    __syncthreads();

    // ---- Compute: 4 output pixels per thread ----------------------------
    // Thread t covers output columns [4t, 4t+4) -> input columns [8t, 8t+8)
    // of both staged rows: four ds_load_b128, one global_store_b128.
    const f4* __restrict__ r0 = (const f4*)(&lds[0]);
    const f4* __restrict__ r1 = (const f4*)(&lds[IN_W]);

    const f4 a0 = r0[2 * t];
    const f4 a1 = r0[2 * t + 1];
    const f4 b0 = r1[2 * t];
    const f4 b1 = r1[2 * t + 1];

    f4 o;
    o.x = rintf(0.25f * ((a0.x + a0.y) + (b0.x + b0.y)));
    o.y = rintf(0.25f * ((a0.z + a0.w) + (b0.z + b0.w)));
    o.z = rintf(0.25f * ((a1.x + a1.y) + (b1.x + b1.y)));
    o.w = rintf(0.25f * ((a1.z + a1.w) + (b1.z + b1.w)));

    // reference casts to uint8; inputs are in [0,255] so this is a no-op
    // clamp kept for safety
    o.x = fminf(fmaxf(o.x, 0.0f), 255.0f);
    o.y = fminf(fmaxf(o.y, 0.0f), 255.0f);
    o.z = fminf(fmaxf(o.z, 0.0f), 255.0f);
    o.w = fminf(fmaxf(o.w, 0.0f), 255.0f);

    f4* __restrict__ outv = (f4*)(dst + ((size_t)nc * OUT_H + oy) * OUT_W);
    outv[t] = o;
}

extern "C" void kernel_launch(void* const* d_in, const int* in_sizes, int n_in,
                              void* d_out, int out_size, void* d_ws, size_t ws_size,
                              hipStream_t stream) {
    (void)n_in; (void)out_size; (void)d_ws; (void)ws_size;

    // setup_inputs order: arg0, arg1, arg2(H), arg3(img), arg4(out_h), arg5(out_w)
    const float* src = (const float*)d_in[3];
    float*       dst = (float*)d_out;

    // N*C derived from the image's flat element count (8*3 = 24 for the
    // reference shapes).
    const int nc_total = in_sizes[3] / (IN_H * IN_W);

    dim3 grid(OUT_H, nc_total > 0 ? nc_total : 1);
    downsample2x_tdm_kernel<<<grid, 256, 0, stream>>>(src, dst);
}